// DAN_14508399526530
// MI455X (gfx1250) — compile-verified
//
#include <hip/hip_runtime.h>
#include <math.h>

typedef float v2f __attribute__((ext_vector_type(2)));
typedef float v4f __attribute__((ext_vector_type(4)));
typedef float v8f __attribute__((ext_vector_type(8)));

#define WPB 8  // waves per block in GEMM kernels (256 threads, wave32)

// ---------- helpers ----------

// Packed-swizzled LDS layout for the 64x64 weight tile.
// Element (row r, col c) with kstep=r>>2, h=(r>>1)&1, p=r&1 is stored at
//   c*64 + h*32 + ((2*kstep + p + 4*c) & 31)
// so that for a given (col, half) the B pairs for k-steps 2j,2j+1 are 4
// contiguous 16B-aligned dwords -> one ds_load_b128 feeding two WMMAs with
// no register shuffles.  The +4*c rotation spreads lanes of each half over a
// disjoint 32-bank range (64 x 4B banks).
__device__ __forceinline__ int lds_pack(int r, int c) {
    int kstep = r >> 2;
    int h = (r >> 1) & 1;
    int p = r & 1;
    return c * 64 + h * 32 + ((2 * kstep + p + 4 * c) & 31);
}

// One j-iteration of the 16x64 output tile: consumes A pairs a0 (k=2j) and
// a1 (k=2j+1), loads B for 4 column tiles via ds_load_b128, issues 8 WMMAs.
__device__ __forceinline__ void wmma_step(const float* wlds, int lid, int half,
                                          int j, v2f a0, v2f a1, v8f acc[4]) {
    int boff = lid * 64 + half * 32 + ((4 * j + 4 * lid) & 31);
#pragma unroll
    for (int n = 0; n < 4; ++n) {
        v4f b4 = *(const v4f*)(wlds + boff + n * 1024);
        v2f b0 = __builtin_shufflevector(b4, b4, 0, 1);
        v2f b1 = __builtin_shufflevector(b4, b4, 2, 3);
        acc[n] = __builtin_amdgcn_wmma_f32_16x16x4_f32(
            false, a0, false, b0, (short)0, acc[n], false, false);
        acc[n] = __builtin_amdgcn_wmma_f32_16x16x4_f32(
            false, a1, false, b1, (short)0, acc[n], false, false);
    }
}

// float atomic-max via the standard sign-aware int trick (works for mixed
// signs; init must be -inf).
__device__ __forceinline__ void atomic_max_f32(float* addr, float val) {
    if (val >= 0.0f) {
        atomicMax((int*)addr, __float_as_int(val));
    } else {
        atomicMin((unsigned int*)addr, __float_as_uint(val));
    }
}

__global__ void k_fill(float* __restrict__ p, float v, int n) {
    int i = blockIdx.x * blockDim.x + threadIdx.x;
    if (i < n) p[i] = v;
}

// ---------- GEMM kernels: C[16-tile, 64] = A[16-tile, 64] @ W[64, 64] (+bias) ----------
// Per-wave 16x64 output tile: 4 accumulators (v8f), 16 k-steps of
// V_WMMA_F32_16X16X4_F32.  A-fragment layout (ISA 7.12.2, 32-bit A 16x4):
//   lane (lid + 16*half): VGPR0 = A[lid][2*half], VGPR1 = A[lid][2*half+1]
// B 4x16: VGPR0 = B[2*half][lid], VGPR1 = B[2*half+1][lid]
// D 16x16: VGPR jj -> row (jj + 8*half), col lid.

__global__ void k_gemm_in(const float* __restrict__ X, const float* __restrict__ W,
                          const float* __restrict__ bias, float* __restrict__ OutM,
                          int ntiles) {
    __shared__ float wlds[64 * 64];
    for (int i = threadIdx.x; i < 64 * 64; i += blockDim.x)
        wlds[lds_pack(i >> 6, i & 63)] = W[i];
    __syncthreads();

    int wave = threadIdx.x >> 5;
    int lane = threadIdx.x & 31;
    int half = lane >> 4, lid = lane & 15;
    int tile = blockIdx.x * WPB + wave;
    if (tile >= ntiles) return;

    const float* arow = X + ((size_t)tile * 16 + lid) * 64 + 2 * half;
    v8f acc[4];
#pragma unroll
    for (int n = 0; n < 4; ++n) acc[n] = (v8f)(0.0f);

#pragma unroll
    for (int j = 0; j < 8; ++j) {
        v2f a0 = *(const v2f*)(arow + 8 * j);
        v2f a1 = *(const v2f*)(arow + 8 * j + 4);
        wmma_step(wlds, lid, half, j, a0, a1, acc);
    }
#pragma unroll
    for (int n = 0; n < 4; ++n) {
        float bn = bias ? bias[n * 16 + lid] : 0.0f;
#pragma unroll
        for (int jj = 0; jj < 8; ++jj) {
            int row = tile * 16 + jj + half * 8;
            OutM[(size_t)row * 64 + n * 16 + lid] = acc[n][jj] + bn;
        }
    }
}

// y[r] = x @ W_rel[r], one relation per blockIdx.y (16KB LDS each)
__global__ void k_gemm_rel(const float* __restrict__ X, const float* __restrict__ Wrel,
                           float* __restrict__ Y, int ntiles, int N) {
    const float* W = Wrel + (size_t)blockIdx.y * 64 * 64;
    float* OutM = Y + (size_t)blockIdx.y * N * 64;

    __shared__ float wlds[64 * 64];
    for (int i = threadIdx.x; i < 64 * 64; i += blockDim.x)
        wlds[lds_pack(i >> 6, i & 63)] = W[i];
    __syncthreads();

    int wave = threadIdx.x >> 5;
    int lane = threadIdx.x & 31;
    int half = lane >> 4, lid = lane & 15;
    int tile = blockIdx.x * WPB + wave;
    if (tile >= ntiles) return;

    const float* arow = X + ((size_t)tile * 16 + lid) * 64 + 2 * half;
    v8f acc[4];
#pragma unroll
    for (int n = 0; n < 4; ++n) acc[n] = (v8f)(0.0f);

#pragma unroll
    for (int j = 0; j < 8; ++j) {
        v2f a0 = *(const v2f*)(arow + 8 * j);
        v2f a1 = *(const v2f*)(arow + 8 * j + 4);
        wmma_step(wlds, lid, half, j, a0, a1, acc);
    }
#pragma unroll
    for (int n = 0; n < 4; ++n) {
#pragma unroll
        for (int jj = 0; jj < 8; ++jj) {
            int row = tile * 16 + jj + half * 8;
            OutM[(size_t)row * 64 + n * 16 + lid] = acc[n][jj];
        }
    }
}

// out = (gcn + 0.5*relu(msg)) @ W_out + b_out
__global__ void k_gemm_out(const float* __restrict__ Gcn, const float* __restrict__ Msg,
                           const float* __restrict__ W, const float* __restrict__ bias,
                           float* __restrict__ OutM, int ntiles) {
    __shared__ float wlds[64 * 64];
    for (int i = threadIdx.x; i < 64 * 64; i += blockDim.x)
        wlds[lds_pack(i >> 6, i & 63)] = W[i];
    __syncthreads();

    int wave = threadIdx.x >> 5;
    int lane = threadIdx.x & 31;
    int half = lane >> 4, lid = lane & 15;
    int tile = blockIdx.x * WPB + wave;
    if (tile >= ntiles) return;

    size_t rbase = ((size_t)tile * 16 + lid) * 64 + 2 * half;
    v8f acc[4];
#pragma unroll
    for (int n = 0; n < 4; ++n) acc[n] = (v8f)(0.0f);

#pragma unroll
    for (int j = 0; j < 8; ++j) {
        v2f g0 = *(const v2f*)(Gcn + rbase + 8 * j);
        v2f m0 = *(const v2f*)(Msg + rbase + 8 * j);
        v2f g1 = *(const v2f*)(Gcn + rbase + 8 * j + 4);
        v2f m1 = *(const v2f*)(Msg + rbase + 8 * j + 4);
        v2f a0, a1;
        a0.x = g0.x + 0.5f * fmaxf(m0.x, 0.0f);
        a0.y = g0.y + 0.5f * fmaxf(m0.y, 0.0f);
        a1.x = g1.x + 0.5f * fmaxf(m1.x, 0.0f);
        a1.y = g1.y + 0.5f * fmaxf(m1.y, 0.0f);
        wmma_step(wlds, lid, half, j, a0, a1, acc);
    }
#pragma unroll
    for (int n = 0; n < 4; ++n) {
        float bn = bias[n * 16 + lid];
#pragma unroll
        for (int jj = 0; jj < 8; ++jj) {
            int row = tile * 16 + jj + half * 8;
            OutM[(size_t)row * 64 + n * 16 + lid] = acc[n][jj] + bn;
        }
    }
}

// ---------- edge-side kernels (gathers from L2-resident x / y) ----------

__global__ void k_deg(const int* __restrict__ eidx, float* __restrict__ deg, int E) {
    int e = blockIdx.x * blockDim.x + threadIdx.x;
    if (e < E) atomicAdd(&deg[eidx[E + e]], 1.0f);   // col = edge_index[1]
}

__global__ void k_dis(const float* __restrict__ deg, float* __restrict__ dis, int N) {
    int n = blockIdx.x * blockDim.x + threadIdx.x;
    if (n < N) {
        float d = deg[n];
        dis[n] = (d > 0.0f) ? rsqrtf(fmaxf(d, 1.0f)) : 0.0f;
    }
}

// pass 1: segment max of res over destination rows.  res[e] = y[type][col].
// 16 threads per edge, float4 per thread.
__global__ void k_edge_max(const float* __restrict__ Y, const int* __restrict__ eidx,
                           const int* __restrict__ etype, float* __restrict__ smax,
                           int E, int N) {
    int idx = blockIdx.x * blockDim.x + threadIdx.x;
    if (idx >= E * 16) return;
    int e = idx >> 4;
    int h = (idx & 15) * 4;
    int r = eidx[e], c = eidx[E + e], t = etype[e];
    float4 v = *(const float4*)(Y + ((size_t)t * N + c) * 64 + h);
    float* p = smax + (size_t)r * 64 + h;
    atomic_max_f32(p + 0, v.x);
    atomic_max_f32(p + 1, v.y);
    atomic_max_f32(p + 2, v.z);
    atomic_max_f32(p + 3, v.w);
}

// pass 2: exp-sum per row, plus the GCN-normalized segment sum of x[col]
__global__ void k_edge_expsum(const float* __restrict__ Y, const float* __restrict__ X,
                              const int* __restrict__ eidx, const int* __restrict__ etype,
                              const float* __restrict__ smax, const float* __restrict__ dis,
                              float* __restrict__ ssum, float* __restrict__ gcn,
                              int E, int N) {
    int idx = blockIdx.x * blockDim.x + threadIdx.x;
    if (idx >= E * 16) return;
    int e = idx >> 4;
    int h = (idx & 15) * 4;
    int r = eidx[e], c = eidx[E + e], t = etype[e];
    float nw = dis[r] * dis[c];
    float4 v  = *(const float4*)(Y + ((size_t)t * N + c) * 64 + h);
    float4 xv = *(const float4*)(X + (size_t)c * 64 + h);
    const float* sm = smax + (size_t)r * 64 + h;
    float* ps = ssum + (size_t)r * 64 + h;
    float* pg = gcn  + (size_t)r * 64 + h;
    atomicAdd(ps + 0, expf(v.x - sm[0]));
    atomicAdd(ps + 1, expf(v.y - sm[1]));
    atomicAdd(ps + 2, expf(v.z - sm[2]));
    atomicAdd(ps + 3, expf(v.w - sm[3]));
    atomicAdd(pg + 0, xv.x * nw);
    atomicAdd(pg + 1, xv.y * nw);
    atomicAdd(pg + 2, xv.z * nw);
    atomicAdd(pg + 3, xv.w * nw);
}

// pass 3: msg[row] += res * exp(res - smax) / (ssum + 1e-16)
__global__ void k_edge_msg(const float* __restrict__ Y, const int* __restrict__ eidx,
                           const int* __restrict__ etype, const float* __restrict__ smax,
                           const float* __restrict__ ssum, float* __restrict__ msg,
                           int E, int N) {
    int idx = blockIdx.x * blockDim.x + threadIdx.x;
    if (idx >= E * 16) return;
    int e = idx >> 4;
    int h = (idx & 15) * 4;
    int r = eidx[e], c = eidx[E + e], t = etype[e];
    float4 v = *(const float4*)(Y + ((size_t)t * N + c) * 64 + h);
    const float* sm = smax + (size_t)r * 64 + h;
    const float* ps = ssum + (size_t)r * 64 + h;
    float* pm = msg + (size_t)r * 64 + h;
    atomicAdd(pm + 0, v.x * (expf(v.x - sm[0]) / (ps[0] + 1e-16f)));
    atomicAdd(pm + 1, v.y * (expf(v.y - sm[1]) / (ps[1] + 1e-16f)));
    atomicAdd(pm + 2, v.z * (expf(v.z - sm[2]) / (ps[2] + 1e-16f)));
    atomicAdd(pm + 3, v.w * (expf(v.w - sm[3]) / (ps[3] + 1e-16f)));
}

// ---------- launcher ----------

extern "C" void kernel_launch(void* const* d_in, const int* in_sizes, int n_in,
                              void* d_out, int out_size, void* d_ws, size_t ws_size,
                              hipStream_t stream) {
    const float* risk = (const float*)d_in[0];
    // d_in[1] = edge_weight (unused by reference)
    const float* Win  = (const float*)d_in[2];
    const float* binw = (const float*)d_in[3];
    const float* Wrel = (const float*)d_in[4];
    const float* Wout = (const float*)d_in[5];
    const float* bout = (const float*)d_in[6];
    const int*   eidx = (const int*)d_in[7];
    const int*   etyp = (const int*)d_in[8];

    const int N = in_sizes[0] / 64;          // 50000
    const int E = in_sizes[8];               // 800000
    const int R = in_sizes[4] / (64 * 64);   // 5
    const size_t NH = (size_t)N * 64;

    float* ws   = (float*)d_ws;
    float* x    = ws;                        // [N,64]
    float* y    = x + NH;                    // [R,N,64]
    float* deg  = y + (size_t)R * NH;        // [N]
    float* dis  = deg + N;                   // [N]
    float* smax = dis + N;                   // [N,64]
    float* ssum = smax + NH;                 // [N,64]
    float* gcn  = ssum + NH;                 // [N,64]
    float* msg  = gcn + NH;                  // [N,64]

    const int ntiles = (N + 15) / 16;
    const dim3 gemm_grid((ntiles + WPB - 1) / WPB);
    const dim3 gemm_blk(WPB * 32);
    const int edge_threads = E * 16;
    const int edge_blocks = (edge_threads + 255) / 256;

    // init accumulators
    k_fill<<<(N + 255) / 256, 256, 0, stream>>>(deg, 0.0f, N);
    k_fill<<<((int)NH + 255) / 256, 256, 0, stream>>>(smax, -INFINITY, (int)NH);
    k_fill<<<((int)(3 * NH) + 255) / 256, 256, 0, stream>>>(ssum, 0.0f, (int)(3 * NH));

    // x = risk @ W_in + b_in
    k_gemm_in<<<gemm_grid, gemm_blk, 0, stream>>>(risk, Win, binw, x, ntiles);

    // y[r] = x @ W_rel[r]  (dense, L2-resident afterwards)
    k_gemm_rel<<<dim3(gemm_grid.x, R), gemm_blk, 0, stream>>>(x, Wrel, y, ntiles, N);

    // degree / inverse sqrt
    k_deg<<<(E + 255) / 256, 256, 0, stream>>>(eidx, deg, E);
    k_dis<<<(N + 255) / 256, 256, 0, stream>>>(deg, dis, N);

    // segment softmax over rows + GCN message
    k_edge_max<<<edge_blocks, 256, 0, stream>>>(y, eidx, etyp, smax, E, N);
    k_edge_expsum<<<edge_blocks, 256, 0, stream>>>(y, x, eidx, etyp, smax, dis,
                                                   ssum, gcn, E, N);
    k_edge_msg<<<edge_blocks, 256, 0, stream>>>(y, eidx, etyp, smax, ssum, msg, E, N);

    // out = (gcn + 0.5*relu(msg)) @ W_out + b_out
    k_gemm_out<<<gemm_grid, gemm_blk, 0, stream>>>(gcn, msg, Wout, bout,
                                                   (float*)d_out, ntiles);
}